// KNNAttention_63702954934814
// MI455X (gfx1250) — compile-verified
//
#include <hip/hip_runtime.h>
#include <hip/hip_bf16.h>

// ---------------- problem constants (reference shapes are fixed) -----------
constexpr int Bc   = 2;
constexpr int Tc   = 2048;
constexpr int Cc   = 768;
constexpr int Hc   = 12;
constexpr int Dc   = 64;        // head dim
constexpr int NMEM = 16384;
constexpr int C3   = 3 * Cc;

typedef __attribute__((ext_vector_type(2))) float v2f;
typedef __attribute__((ext_vector_type(8))) float v8f;

#define WMMA_F32(a, b, c) \
    __builtin_amdgcn_wmma_f32_16x16x4_f32(false, (a), false, (b), (short)0, (c), false, false)

// ---------------------------------------------------------------------------
// Register-blocked fp32 WMMA GEMM: Cm[M,N] = A[M,K] * Bm[K,N] + bias[N].
// One wave computes a 64x32 macro-tile = 4x2 tiles of 16x16, K stepped by 4
// through V_WMMA_F32_16X16X4_F32. Per k-step: 4 A-frags + 2 B-frags feed 8
// WMMAs. 8 accumulators = 64 VGPRs -> no spills (4x4 blocking spilled).
// Fragment layouts (ISA 7.12.2):
//   A-frag (16x4): lane l, reg r -> A[m0 + (l&15)][k0 + 2*(l>>4) + r]
//   B-frag (4x16): lane l, reg r -> Bm[k0 + 2*(l>>4) + r][n0 + (l&15)]
//   C/D (16x16)  : lane l, reg r -> Cm[m0 + r + 8*(l>>4)][n0 + (l&15)]
// Requires M % 64 == 0, N % 32 == 0, K % 4 == 0.
// ---------------------------------------------------------------------------
__global__ void __launch_bounds__(32)
gemm_wmma_f32_4x2(const float* __restrict__ A,
                  const float* __restrict__ Bm,
                  const float* __restrict__ bias,
                  float* __restrict__ Cm,
                  int M, int N, int K)
{
    const int n0 = blockIdx.x * 32;
    const int m0 = blockIdx.y * 64;
    const int l  = threadIdx.x;
    const int lo = l & 15;
    const int hi = l >> 4;

    v8f acc[4][2] = {};

    const float* aptr = A  + (size_t)(m0 + lo) * K + 2 * hi;   // + mt*16*K
    const float* bptr = Bm + (size_t)(2 * hi) * N + n0 + lo;   // + nt*16

    const size_t aStride = (size_t)16 * K;

    for (int k0 = 0; k0 < K; k0 += 4) {
        v2f a0 = *(const v2f*)(aptr);
        v2f a1 = *(const v2f*)(aptr + aStride);
        v2f a2 = *(const v2f*)(aptr + 2 * aStride);
        v2f a3 = *(const v2f*)(aptr + 3 * aStride);
        v2f b0, b1;
        b0[0] = bptr[0];
        b0[1] = bptr[N];
        b1[0] = bptr[16];
        b1[1] = bptr[16 + N];

        acc[0][0] = WMMA_F32(a0, b0, acc[0][0]);
        acc[0][1] = WMMA_F32(a0, b1, acc[0][1]);
        acc[1][0] = WMMA_F32(a1, b0, acc[1][0]);
        acc[1][1] = WMMA_F32(a1, b1, acc[1][1]);
        acc[2][0] = WMMA_F32(a2, b0, acc[2][0]);
        acc[2][1] = WMMA_F32(a2, b1, acc[2][1]);
        acc[3][0] = WMMA_F32(a3, b0, acc[3][0]);
        acc[3][1] = WMMA_F32(a3, b1, acc[3][1]);

        aptr += 4;
        bptr += (size_t)4 * N;
    }

#pragma unroll
    for (int nt = 0; nt < 2; ++nt) {
        const float bn = bias[n0 + nt * 16 + lo];
#pragma unroll
        for (int mt = 0; mt < 4; ++mt) {
            float* cptr = Cm + (size_t)(m0 + mt * 16 + 8 * hi) * N
                             + n0 + nt * 16 + lo;
#pragma unroll
            for (int r = 0; r < 8; ++r)
                cptr[(size_t)r * N] = acc[mt][nt][r] + bn;
        }
    }
}

// ---------------------------------------------------------------------------
// Flash attention, one wave per (b,h, 16-row query block).
// qkv layout: [B, T, 3C] with q at +0, k at +C, v at +2C, head offset h*D.
// Output y written as [B, T, C] (head-major already transposed).
// ---------------------------------------------------------------------------
__global__ void __launch_bounds__(32)
flash_attn_kernel(const float* __restrict__ qkv,
                  float* __restrict__ y)
{
    const int qblk = blockIdx.x;          // T/16
    const int bh   = blockIdx.y;          // B*H
    const int b    = bh / Hc;
    const int h    = bh % Hc;
    const int l    = threadIdx.x;
    const int lo   = l & 15;
    const int hi   = l >> 4;

    __shared__ float sS[16 * 16];
    __shared__ float sAl[16];

    const float* qbase = qkv + (size_t)(b * Tc) * C3 + h * Dc;
    const float* kbase = qbase + Cc;
    const float* vbase = qbase + 2 * Cc;

    const int m0 = qblk * 16;

    // Preload Q A-fragments for all 16 K-steps (K = d = 64).
    v2f qA[16];
    {
        const float* qrow = qbase + (size_t)(m0 + lo) * C3 + 2 * hi;
#pragma unroll
        for (int kk = 0; kk < 16; ++kk)
            qA[kk] = *(const v2f*)(qrow + 4 * kk);
    }

    v8f  O[4] = {};                       // 4 d-tiles of 16 cols each
    float m_i = -1e30f, l_i = 0.0f;       // row stats (meaningful in lanes 0..15)

    for (int sblk = 0; sblk <= qblk; ++sblk) {
        const int s0 = sblk * 16;

        // ---- S = Q K^T for this 16x16 tile ------------------------------
        v8f s = {};
        const float* krow = kbase + (size_t)(s0 + lo) * C3 + 2 * hi;
#pragma unroll
        for (int kk = 0; kk < 16; ++kk) {
            v2f kb = *(const v2f*)(krow + 4 * kk);
            s = WMMA_F32(qA[kk], kb, s);
        }

        // ---- scale + causal mask, spill tile to LDS ---------------------
        const bool diag = (sblk == qblk);
#pragma unroll
        for (int r = 0; r < 8; ++r) {
            const int m  = r + 8 * hi;            // row within tile
            float val = s[r] * 0.125f;            // 1/sqrt(64)
            if (diag && (lo > m)) val = -1e30f;   // key col > query row
            sS[m * 16 + lo] = val;
        }
        __syncthreads();

        // ---- online softmax, one lane per row ---------------------------
        if (l < 16) {
            float rowmax = -1e30f;
#pragma unroll
            for (int j = 0; j < 16; ++j) rowmax = fmaxf(rowmax, sS[l * 16 + j]);
            const float m_new = fmaxf(m_i, rowmax);
            const float alpha = __expf(m_i - m_new);
            float ssum = 0.0f;
#pragma unroll
            for (int j = 0; j < 16; ++j) {
                const float p = __expf(sS[l * 16 + j] - m_new);
                sS[l * 16 + j] = p;
                ssum += p;
            }
            l_i = alpha * l_i + ssum;
            m_i = m_new;
            sAl[l] = alpha;
        }
        __syncthreads();

        // ---- rescale accumulator by per-row alpha -----------------------
#pragma unroll
        for (int r = 0; r < 8; ++r) {
            const float a = sAl[r + 8 * hi];
#pragma unroll
            for (int dt = 0; dt < 4; ++dt) O[dt][r] *= a;
        }

        // ---- O += P x V  (P re-read from LDS in A-fragment layout) ------
#pragma unroll
        for (int kk = 0; kk < 4; ++kk) {
            v2f pa = *(const v2f*)(&sS[lo * 16 + 4 * kk + 2 * hi]);
            const float* vrow = vbase + (size_t)(s0 + 4 * kk + 2 * hi) * C3 + lo;
#pragma unroll
            for (int dt = 0; dt < 4; ++dt) {
                v2f vb;
                vb[0] = vrow[dt * 16];
                vb[1] = vrow[dt * 16 + (size_t)C3];
                O[dt] = WMMA_F32(pa, vb, O[dt]);
            }
        }
        __syncthreads();   // sS reused next iteration
    }

    // ---- normalize by row sum and store y[B,T,C] ------------------------
    if (l < 16) sAl[l] = 1.0f / l_i;
    __syncthreads();

    float* ybase = y + (size_t)(b * Tc) * Cc + h * Dc;
#pragma unroll
    for (int r = 0; r < 8; ++r) {
        const int   m   = r + 8 * hi;
        const float inv = sAl[m];
        float* yrow = ybase + (size_t)(m0 + m) * Cc;
#pragma unroll
        for (int dt = 0; dt < 4; ++dt)
            yrow[dt * 16 + lo] = O[dt][r] * inv;
    }
}

// ---------------------------------------------------------------------------
// KNN memory attention + gate blend. One wave per (b,t,h); each lane owns
// 2 of the 64 head-dim elements.
// ---------------------------------------------------------------------------
__global__ void knn_mem_kernel(const float* __restrict__ qkv,
                               const float* __restrict__ y,
                               const float* __restrict__ mem_bank,
                               const int*   __restrict__ knn_idx,
                               const float* __restrict__ gate_bias,
                               float* __restrict__ combined)
{
    const int wid  = blockIdx.x * (blockDim.x >> 5) + (threadIdx.x >> 5);
    const int lane = threadIdx.x & 31;
    const int h    = wid % Hc;
    const int bt   = wid / Hc;            // b*T + t
    const int b    = bt / Tc;

    const float2 qv = *(const float2*)(qkv + (size_t)bt * C3 + h * Dc + lane * 2);

    const int idxbase = bt * 3;
    float sc[3];
#pragma unroll
    for (int kk = 0; kk < 3; ++kk) {
        const int idx = knn_idx[idxbase + kk];
        const float* mk = mem_bank
            + ((size_t)(b * NMEM + idx) * 2 + 0) * Cc + h * Dc + lane * 2;
        const float2 mkv = *(const float2*)mk;
        float part = qv.x * mkv.x + qv.y * mkv.y;
#pragma unroll
        for (int off = 16; off > 0; off >>= 1)
            part += __shfl_xor(part, off, 32);
        sc[kk] = part * 0.125f;
    }

    const float mx = fmaxf(sc[0], fmaxf(sc[1], sc[2]));
    const float e0 = __expf(sc[0] - mx);
    const float e1 = __expf(sc[1] - mx);
    const float e2 = __expf(sc[2] - mx);
    const float inv = 1.0f / (e0 + e1 + e2);
    const float p[3] = { e0 * inv, e1 * inv, e2 * inv };

    float2 acc = make_float2(0.0f, 0.0f);
#pragma unroll
    for (int kk = 0; kk < 3; ++kk) {
        const int idx = knn_idx[idxbase + kk];
        const float* mv = mem_bank
            + ((size_t)(b * NMEM + idx) * 2 + 1) * Cc + h * Dc + lane * 2;
        const float2 mvv = *(const float2*)mv;
        acc.x += p[kk] * mvv.x;
        acc.y += p[kk] * mvv.y;
    }

    const float g = gate_bias[h];
    const float2 yv = *(const float2*)(y + (size_t)bt * Cc + h * Dc + lane * 2);
    float2 outv;
    outv.x = acc.x * g + yv.x * (1.0f - g);
    outv.y = acc.y * g + yv.y * (1.0f - g);
    *(float2*)(combined + (size_t)bt * Cc + h * Dc + lane * 2) = outv;
}

// ---------------------------------------------------------------------------
// kv_memories = stack(k, v) -> [B, T, 2, C]
// ---------------------------------------------------------------------------
__global__ void kv_copy_kernel(const float* __restrict__ qkv,
                               float* __restrict__ kv_out)
{
    const size_t i = (size_t)blockIdx.x * blockDim.x + threadIdx.x;
    const size_t total = (size_t)Bc * Tc * 2 * Cc;
    if (i >= total) return;
    const size_t c     = i % Cc;
    const size_t rest  = i / Cc;
    const size_t which = rest % 2;        // 0 -> k, 1 -> v
    const size_t bt    = rest / 2;
    kv_out[i] = qkv[bt * C3 + (1 + which) * Cc + c];
}

// ---------------------------------------------------------------------------
extern "C" void kernel_launch(void* const* d_in, const int* in_sizes, int n_in,
                              void* d_out, int out_size, void* d_ws, size_t ws_size,
                              hipStream_t stream)
{
    const float* x         = (const float*)d_in[0];
    const float* mem_bank  = (const float*)d_in[1];
    const float* W_attn    = (const float*)d_in[2];
    const float* b_attn    = (const float*)d_in[3];
    const float* W_proj    = (const float*)d_in[4];
    const float* b_proj    = (const float*)d_in[5];
    const float* gate_bias = (const float*)d_in[6];
    const int*   knn_idx   = (const int*)d_in[7];

    float* out    = (float*)d_out;                     // [B,T,C]
    float* kv_out = out + (size_t)Bc * Tc * Cc;        // [B,T,2,C]

    float* qkv  = (float*)d_ws;                        // [B,T,3C]
    float* yws  = qkv + (size_t)Bc * Tc * C3;          // [B,T,C]
    float* comb = yws + (size_t)Bc * Tc * Cc;          // [B,T,C]

    // 1) qkv = x @ W_attn + b_attn   (4096 x 768 -> 2304)
    {
        dim3 grid(C3 / 32, (Bc * Tc) / 64);            // (72, 64)
        gemm_wmma_f32_4x2<<<grid, 32, 0, stream>>>(x, W_attn, b_attn, qkv,
                                                   Bc * Tc, C3, Cc);
    }
    // 2) causal flash attention -> y [B,T,C]
    {
        dim3 grid(Tc / 16, Bc * Hc);                   // (128, 24)
        flash_attn_kernel<<<grid, 32, 0, stream>>>(qkv, yws);
    }
    // 3) KNN memory attention + gate blend -> combined [B,T,C]
    {
        const int waves  = Bc * Tc * Hc;               // 49152
        const int blocks = waves / 8;                  // 256 threads = 8 waves
        knn_mem_kernel<<<blocks, 256, 0, stream>>>(qkv, yws, mem_bank, knn_idx,
                                                   gate_bias, comb);
    }
    // 4) out = combined @ W_proj + b_proj   (4096 x 768 -> 768)
    {
        dim3 grid(Cc / 32, (Bc * Tc) / 64);            // (24, 64)
        gemm_wmma_f32_4x2<<<grid, 32, 0, stream>>>(comb, W_proj, b_proj, out,
                                                   Bc * Tc, Cc, Cc);
    }
    // 5) kv_memories
    {
        const size_t total = (size_t)Bc * Tc * 2 * Cc;
        kv_copy_kernel<<<(unsigned)((total + 255) / 256), 256, 0, stream>>>(qkv, kv_out);
    }
}